// CrossCorrelation_74758200754433
// MI455X (gfx1250) — compile-verified
//
#include <hip/hip_runtime.h>
#include <math.h>

// Native ext-vector types (WMMA operand shapes + 128-bit load payloads).
typedef __attribute__((ext_vector_type(2))) float v2f;
typedef __attribute__((ext_vector_type(4))) float f32x4;
typedef __attribute__((ext_vector_type(8))) float v8f;

// Wave32-wide sum via V_WMMA_F32_16X16X4_F32.
// A per-lane = {v, 0}: documented A layout (16x4 f32) gives
//   A[m][0] = v_m          (lanes 0..15, VGPR0)
//   A[m][2] = v_{m+16}     (lanes 16..31, VGPR0)
//   A[m][1] = A[m][3] = 0
// B = all ones (layout-independent).  =>  D[m][n] = v_m + v_{m+16}.
// Documented C/D layout: d[r] in lanes 0..15 holds row r, lanes 16..31 row r+8.
// Summing d[0..7] in-lane gives rows 0..7 (lanes 0-15) / rows 8..15 (lanes 16-31);
// one shfl_xor(16) completes the full 32-lane sum in every lane.
__device__ __forceinline__ float wave_sum32(float v) {
    v2f a; a[0] = v;    a[1] = 0.0f;
    v2f b; b[0] = 1.0f; b[1] = 1.0f;
    v8f c = {0.f, 0.f, 0.f, 0.f, 0.f, 0.f, 0.f, 0.f};
    v8f d = __builtin_amdgcn_wmma_f32_16x16x4_f32(
        /*neg_a=*/false, a, /*neg_b=*/false, b,
        /*c_mod=*/(short)0, c, /*reuse_a=*/false, /*reuse_b=*/false);
    float s = ((d[0] + d[1]) + (d[2] + d[3])) + ((d[4] + d[5]) + (d[6] + d[7]));
    s += __shfl_xor(s, 16, 32);
    return s;
}

// Pass 1: streaming partial reduction. 256 threads = 8 wave32 per block.
__global__ void __launch_bounds__(256) wpearson_partial_kernel(
    const float* __restrict__ xs,
    const float* __restrict__ ys,
    const float* __restrict__ ns,
    float* __restrict__ partials,
    int n, int n4)
{
    float s_n = 0.f, s_x = 0.f, s_y = 0.f, s_x2 = 0.f, s_y2 = 0.f, s_xy = 0.f;

    const f32x4* __restrict__ xv4 = reinterpret_cast<const f32x4*>(xs);
    const f32x4* __restrict__ yv4 = reinterpret_cast<const f32x4*>(ys);
    const f32x4* __restrict__ nv4 = reinterpret_cast<const f32x4*>(ns);

    const unsigned stride = gridDim.x * blockDim.x;
    for (unsigned i = blockIdx.x * blockDim.x + threadIdx.x; i < (unsigned)n4; i += stride) {
        // Streamed once, working set > L2: non-temporal 128-bit loads.
        f32x4 xv = __builtin_nontemporal_load(xv4 + i);
        f32x4 yv = __builtin_nontemporal_load(yv4 + i);
        f32x4 nv = __builtin_nontemporal_load(nv4 + i);
#pragma unroll
        for (int k = 0; k < 4; ++k) {
            float xx = xv[k], yy = yv[k], nn = nv[k];
            float nx = nn * xx;
            float ny = nn * yy;
            s_n  += nn;
            s_x  += nx;
            s_y  += ny;
            s_x2 = fmaf(nx, xx, s_x2);
            s_y2 = fmaf(ny, yy, s_y2);
            s_xy = fmaf(nx, yy, s_xy);
        }
    }

    // Scalar tail (n % 4) — tiny; fully reconverges before the WMMA reduction.
    if (blockIdx.x == 0 && threadIdx.x == 0) {
        for (int i = n4 * 4; i < n; ++i) {
            float xx = xs[i], yy = ys[i], nn = ns[i];
            float nx = nn * xx, ny = nn * yy;
            s_n += nn; s_x += nx; s_y += ny;
            s_x2 = fmaf(nx, xx, s_x2);
            s_y2 = fmaf(ny, yy, s_y2);
            s_xy = fmaf(nx, yy, s_xy);
        }
    }

    // Wave-wide reduction via WMMA (uniform control flow -> EXEC all ones).
    float acc[6] = { s_n, s_x, s_y, s_x2, s_y2, s_xy };
#pragma unroll
    for (int j = 0; j < 6; ++j) acc[j] = wave_sum32(acc[j]);

    __shared__ float lds[8][6];
    const unsigned lane = threadIdx.x & 31u;
    const unsigned wave = threadIdx.x >> 5;
    if (lane == 0) {
#pragma unroll
        for (int j = 0; j < 6; ++j) lds[wave][j] = acc[j];
    }
    __syncthreads();
    if (threadIdx.x == 0) {
#pragma unroll
        for (int j = 0; j < 6; ++j) {
            float t = 0.f;
#pragma unroll
            for (int w = 0; w < 8; ++w) t += lds[w][j];
            partials[blockIdx.x * 6 + j] = t;
        }
    }
}

// Pass 2: single-block deterministic finalize + scalar formula.
__global__ void __launch_bounds__(256) wpearson_final_kernel(
    const float* __restrict__ partials, int nparts, float* __restrict__ out)
{
    float acc[6] = { 0.f, 0.f, 0.f, 0.f, 0.f, 0.f };
    for (int i = threadIdx.x; i < nparts; i += 256) {
#pragma unroll
        for (int j = 0; j < 6; ++j) acc[j] += partials[i * 6 + j];
    }

#pragma unroll
    for (int j = 0; j < 6; ++j) acc[j] = wave_sum32(acc[j]);

    __shared__ float lds[8][6];
    const unsigned lane = threadIdx.x & 31u;
    const unsigned wave = threadIdx.x >> 5;
    if (lane == 0) {
#pragma unroll
        for (int j = 0; j < 6; ++j) lds[wave][j] = acc[j];
    }
    __syncthreads();
    if (threadIdx.x == 0) {
        float s[6];
#pragma unroll
        for (int j = 0; j < 6; ++j) {
            float t = 0.f;
#pragma unroll
            for (int w = 0; w < 8; ++w) t += lds[w][j];
            s[j] = t;
        }
        const float sum_n = s[0], sum_x = s[1], sum_y = s[2];
        const float sum_x2 = s[3], sum_y2 = s[4], sum_xy = s[5];
        const float num = sum_n * sum_xy - sum_x * sum_y;
        const float den = sqrtf(sum_n * sum_x2 - sum_x * sum_x) *
                          sqrtf(sum_n * sum_y2 - sum_y * sum_y);
        out[0] = num / den;
    }
}

extern "C" void kernel_launch(void* const* d_in, const int* in_sizes, int n_in,
                              void* d_out, int out_size, void* d_ws, size_t ws_size,
                              hipStream_t stream) {
    const float* xs = (const float*)d_in[0];
    const float* ys = (const float*)d_in[1];
    const float* ns = (const float*)d_in[2];
    float* out      = (float*)d_out;
    float* partials = (float*)d_ws;

    const int n  = in_sizes[0];
    const int n4 = n >> 2;

    // Enough blocks to saturate HBM across all WGPs; each thread then does
    // ~16 float4 iterations (n = 2^25 -> 2048 blocks x 256 thr x 16 iters).
    int blocks = 2048;
    int max_by_ws = (int)(ws_size / (6 * sizeof(float)));
    if (blocks > max_by_ws) blocks = max_by_ws;
    int need = (n4 + 255) / 256;
    if (need < 1) need = 1;
    if (blocks > need) blocks = need;
    if (blocks < 1) blocks = 1;

    wpearson_partial_kernel<<<blocks, 256, 0, stream>>>(xs, ys, ns, partials, n, n4);
    wpearson_final_kernel<<<1, 256, 0, stream>>>(partials, blocks, out);
}